// GAT_53772990545978
// MI455X (gfx1250) — compile-verified
//
#include <hip/hip_runtime.h>
#include <hip/hip_bf16.h>
#include <math.h>

typedef __attribute__((ext_vector_type(2))) float v2f;
typedef __attribute__((ext_vector_type(8))) float v8f;

#define N_NODES 100000
#define E_EDGES 1600000
#define IN_DIM  256
#define OUT_DIM 128
#define N_TILES (OUT_DIM / 16)     // 8
#define M_TILES (N_NODES / 16)     // 6250 (exact)
#define LRELU   0.2f

// ---------------- init: zero accumulators / softmax state ----------------
__global__ __launch_bounds__(256) void gat_init(float* __restrict__ out,
                                                unsigned* __restrict__ mkey,
                                                float* __restrict__ ssum) {
    int t = blockIdx.x * blockDim.x + threadIdx.x;
    if (t < N_NODES * OUT_DIM) out[t] = 0.0f;
    if (t < N_NODES) { mkey[t] = 0u; ssum[t] = 0.0f; }
}

// ---------------- GEMM: seq = X @ W via V_WMMA_F32_16X16X4_F32 ----------------
// One wave computes a 16(M) x 128(N) slab: 8 accumulators, K looped in steps of 4.
__global__ __launch_bounds__(256) void gat_gemm(const float* __restrict__ X,
                                                const float* __restrict__ W,
                                                float* __restrict__ seq) {
    int wave = (blockIdx.x * blockDim.x + threadIdx.x) >> 5;
    int lane = threadIdx.x & 31;
    if (wave >= M_TILES) return;            // whole wave exits -> EXEC all-1 for WMMA
    const int m0   = wave * 16;
    const int half = lane >> 4;             // 0: lanes 0-15, 1: lanes 16-31
    const int lid  = lane & 15;

    v8f acc[N_TILES] = {};

    const float* xrow = X + (size_t)(m0 + lid) * IN_DIM;
    for (int k0 = 0; k0 < IN_DIM; k0 += 4) {
        // A fragment (16x4 f32): lanes 0-15 hold K=k0,k0+1; lanes 16-31 hold k0+2,k0+3
        v2f a;
        a.x = xrow[k0 + 2 * half + 0];
        a.y = xrow[k0 + 2 * half + 1];
        #pragma unroll
        for (int t = 0; t < N_TILES; ++t) {
            int n = t * 16 + lid;
            v2f b;                           // B fragment (4x16 f32), mirrored layout
            b.x = W[(size_t)(k0 + 2 * half + 0) * OUT_DIM + n];
            b.y = W[(size_t)(k0 + 2 * half + 1) * OUT_DIM + n];
            acc[t] = __builtin_amdgcn_wmma_f32_16x16x4_f32(
                false, a, false, b, (short)0, acc[t], false, false);
        }
    }
    // C/D layout: VGPR r, lanes 0-15 -> M = r, lanes 16-31 -> M = 8 + r ; N = lid
    #pragma unroll
    for (int t = 0; t < N_TILES; ++t) {
        #pragma unroll
        for (int r = 0; r < 8; ++r) {
            int row = m0 + r + 8 * half;
            int col = t * 16 + lid;
            seq[(size_t)row * OUT_DIM + col] = acc[t][r];
        }
    }
}

// ---------------- per-node attention projections f1, f2 ----------------
__global__ __launch_bounds__(256) void gat_f1f2(const float* __restrict__ seq,
                                                const float* __restrict__ a1,
                                                const float* __restrict__ a2,
                                                const float* __restrict__ b1,
                                                const float* __restrict__ b2,
                                                float* __restrict__ f1,
                                                float* __restrict__ f2) {
    int wave = (blockIdx.x * blockDim.x + threadIdx.x) >> 5;
    int lane = threadIdx.x & 31;
    if (wave >= N_NODES) return;
    const float4* row = (const float4*)(seq + (size_t)wave * OUT_DIM);
    const float4* A1  = (const float4*)a1;
    const float4* A2  = (const float4*)a2;
    float4 v  = row[lane];
    float4 c1 = A1[lane];
    float4 c2 = A2[lane];
    float s1 = v.x * c1.x + v.y * c1.y + v.z * c1.z + v.w * c1.w;
    float s2 = v.x * c2.x + v.y * c2.y + v.z * c2.z + v.w * c2.w;
    #pragma unroll
    for (int off = 16; off > 0; off >>= 1) {
        s1 += __shfl_xor(s1, off, 32);
        s2 += __shfl_xor(s2, off, 32);
    }
    if (lane == 0) {
        f1[wave] = s1 + b1[0];
        f2[wave] = s2 + b2[0];
    }
}

// ---------------- edge logits + segment max (monotone uint key) ----------------
__device__ __forceinline__ unsigned flip_key(float f) {
    unsigned u = __float_as_uint(f);
    return (u & 0x80000000u) ? ~u : (u | 0x80000000u);
}
__device__ __forceinline__ float unflip_key(unsigned k) {
    unsigned u = (k & 0x80000000u) ? (k & 0x7FFFFFFFu) : ~k;
    return __uint_as_float(u);
}

__global__ __launch_bounds__(256) void gat_logit_max(const int* __restrict__ row,
                                                     const int* __restrict__ col,
                                                     const float* __restrict__ f1,
                                                     const float* __restrict__ f2,
                                                     float* __restrict__ logits,
                                                     unsigned* __restrict__ mkey) {
    int e = blockIdx.x * blockDim.x + threadIdx.x;
    if (e >= E_EDGES) return;
    int i = row[e], j = col[e];
    float x = f1[i] + f2[j];
    float l = (x > 0.0f) ? x : LRELU * x;
    logits[e] = l;
    atomicMax(&mkey[i], flip_key(l));
}

// ---------------- exp(logit - max) + segment sum (in-place on logits) ----------------
__global__ __launch_bounds__(256) void gat_expsum(const int* __restrict__ row,
                                                  float* __restrict__ logits,
                                                  const unsigned* __restrict__ mkey,
                                                  float* __restrict__ ssum) {
    int e = blockIdx.x * blockDim.x + threadIdx.x;
    if (e >= E_EDGES) return;
    int i = row[e];
    float m  = unflip_key(mkey[i]);
    float ex = expf(logits[e] - m);
    logits[e] = ex;                          // reuse buffer as e-values
    atomicAdd(&ssum[i], ex);
}

// ---------------- SpMM scatter: one wave per edge, float4 per lane ----------------
__global__ __launch_bounds__(256) void gat_spmm(const int* __restrict__ row,
                                                const int* __restrict__ col,
                                                const float* __restrict__ eexp,
                                                const float* __restrict__ ssum,
                                                const float* __restrict__ seq,
                                                float* __restrict__ out) {
    int wave = (blockIdx.x * blockDim.x + threadIdx.x) >> 5;
    int lane = threadIdx.x & 31;
    if (wave >= E_EDGES) return;
    int i = row[wave], j = col[wave];
    float coef = eexp[wave] / ssum[i];
    float4 v = ((const float4*)(seq + (size_t)j * OUT_DIM))[lane];
    float* dst = out + (size_t)i * OUT_DIM + lane * 4;
    atomicAdd(dst + 0, coef * v.x);
    atomicAdd(dst + 1, coef * v.y);
    atomicAdd(dst + 2, coef * v.z);
    atomicAdd(dst + 3, coef * v.w);
}

// ---------------- finalize: elu(elu(out + bias)) ----------------
__global__ __launch_bounds__(256) void gat_final(float* __restrict__ out,
                                                 const float* __restrict__ bias) {
    int t = blockIdx.x * blockDim.x + threadIdx.x;
    if (t >= N_NODES * OUT_DIM) return;
    float x = out[t] + bias[t & (OUT_DIM - 1)];
    x = (x > 0.0f) ? x : (expf(x) - 1.0f);
    x = (x > 0.0f) ? x : (expf(x) - 1.0f);
    out[t] = x;
}

extern "C" void kernel_launch(void* const* d_in, const int* in_sizes, int n_in,
                              void* d_out, int out_size, void* d_ws, size_t ws_size,
                              hipStream_t stream) {
    const float* X    = (const float*)d_in[0];
    const int*   ei   = (const int*)d_in[1];
    const float* W    = (const float*)d_in[2];
    const float* a1   = (const float*)d_in[3];
    const float* b1   = (const float*)d_in[4];
    const float* a2   = (const float*)d_in[5];
    const float* b2   = (const float*)d_in[6];
    const float* bias = (const float*)d_in[7];
    float* out = (float*)d_out;

    const int* row = ei;              // edge_index[0] = destination (softmax row)
    const int* col = ei + E_EDGES;    // edge_index[1] = source

    // workspace carve-out (all 128B aligned, sizes are multiples of 128B)
    char* ws = (char*)d_ws;
    float*    seq    = (float*)ws;    ws += sizeof(float) * (size_t)N_NODES * OUT_DIM; // 51.2 MB
    float*    f1     = (float*)ws;    ws += sizeof(float) * (size_t)N_NODES;
    float*    f2     = (float*)ws;    ws += sizeof(float) * (size_t)N_NODES;
    unsigned* mkey   = (unsigned*)ws; ws += sizeof(unsigned) * (size_t)N_NODES;
    float*    ssum   = (float*)ws;    ws += sizeof(float) * (size_t)N_NODES;
    float*    logits = (float*)ws;    ws += sizeof(float) * (size_t)E_EDGES;           // 6.4 MB

    const int T = 256;
    gat_init<<<(N_NODES * OUT_DIM + T - 1) / T, T, 0, stream>>>(out, mkey, ssum);
    gat_gemm<<<(M_TILES + 7) / 8, T, 0, stream>>>(X, W, seq);
    gat_f1f2<<<(N_NODES + 7) / 8, T, 0, stream>>>(seq, a1, a2, b1, b2, f1, f2);
    gat_logit_max<<<(E_EDGES + T - 1) / T, T, 0, stream>>>(row, col, f1, f2, logits, mkey);
    gat_expsum<<<(E_EDGES + T - 1) / T, T, 0, stream>>>(row, logits, mkey, ssum);
    gat_spmm<<<(E_EDGES + 7) / 8, T, 0, stream>>>(row, col, logits, ssum, seq, out);
    gat_final<<<(N_NODES * OUT_DIM + T - 1) / T, T, 0, stream>>>(out, bias);
}